// Corr_5016521802286
// MI455X (gfx1250) — compile-verified
//
#include <hip/hip_runtime.h>
#include <math.h>

// ---------------------------------------------------------------------------
// Masked-SSIM blend for MI455X (gfx1250), fp32 end-to-end.
//
// Math: windowed correlation gamma is invariant under global affine
// normalization (x-u)/s, so the 8 convolutions run on RAW images:
//   ch0=a ch1=b ch2=c ch3=a^2 ch4=b^2 ch5=c^2 ch6=a*c ch7=b*c
// Reduction scalars (u1,s1,u2,s2,u3,s3) only feed the final combine.
//
// Conv = separable 11-tap Gaussian as Toeplitz matmuls with
// V_WMMA_F32_16X16X4_F32 (full fp32; K padded 26->28).
// Input tiles staged LDS-ward with global_load_async_to_lds_b32 (ASYNCcnt).
// ---------------------------------------------------------------------------

typedef float v2f __attribute__((ext_vector_type(2)));
typedef float v8f __attribute__((ext_vector_type(8)));

#define H_IMG 2048
#define W_IMG 2048
#define PAD 5

#define OT 32                        // 32x32 output tile per workgroup
#define IT 42                        // input region = OT + 2*PAD
#define IN_ROWS 48                   // zero-padded for K-chunk overreach
#define IN_STRIDE 52                 // even (b64-aligned), conflict-free mod 64
#define IN_SZ (IN_ROWS * IN_STRIDE)  // 2496
#define HB_ROWS 48
#define HB_STRIDE 41                 // half-wave bank sets disjoint on reads
#define HB_SZ (HB_ROWS * HB_STRIDE)  // 1968

// Gaussian weight g[d], d in [0,10]; invS = 1/sum = 1/3.7592308 (fp32).
__device__ __forceinline__ float gweight(int d) {
    float x = (float)(d - 5);
    return expf(-x * x * (1.0f / 4.5f)) * 0.26601216f;
}

__device__ __forceinline__ v8f v8f_zero() {
    v8f z = {0.f, 0.f, 0.f, 0.f, 0.f, 0.f, 0.f, 0.f};
    return z;
}

// ---------------------------------------------------------------------------
// Kernel 1: deterministic per-block masked partial sums
// ---------------------------------------------------------------------------
__global__ __launch_bounds__(256)
void reduce_partial(const float* __restrict__ a, const float* __restrict__ b,
                    const float* __restrict__ c, const int* __restrict__ m,
                    float* __restrict__ partial) {
    __shared__ float red[7][256];
    const int t = threadIdx.x;
    const long base = (long)blockIdx.x * 4096;
    float s0 = 0.f, s1 = 0.f, s2 = 0.f, s3 = 0.f, s4 = 0.f, s5 = 0.f, s6 = 0.f;
#pragma unroll 4
    for (int k = 0; k < 16; ++k) {
        long i = base + (long)k * 256 + t;
        __builtin_prefetch(&a[i + 1024], 0, 0);   // global_prefetch_b8
        __builtin_prefetch(&c[i + 1024], 0, 0);
        float mm = (m[i] > 0) ? 1.f : 0.f;
        float av = a[i], bv = b[i], cv = c[i];
        s0 += mm;
        s1 += mm * av; s2 += mm * av * av;
        s3 += mm * bv; s4 += mm * bv * bv;
        s5 += mm * cv; s6 += mm * cv * cv;
    }
    red[0][t] = s0; red[1][t] = s1; red[2][t] = s2; red[3][t] = s3;
    red[4][t] = s4; red[5][t] = s5; red[6][t] = s6;
    __syncthreads();
    for (int off = 128; off > 0; off >>= 1) {
        if (t < off) {
#pragma unroll
            for (int j = 0; j < 7; ++j) red[j][t] += red[j][t + off];
        }
        __syncthreads();
    }
    if (t < 7) partial[(long)blockIdx.x * 7 + t] = red[t][0];
}

// ---------------------------------------------------------------------------
// Kernel 2: final deterministic reduction -> scal[0..5] = u1,s1,u2,s2,u3,s3
// ---------------------------------------------------------------------------
__global__ __launch_bounds__(256)
void reduce_final(const float* __restrict__ partial, float* __restrict__ scal) {
    __shared__ float red[256];
    __shared__ float tot[7];
    const int t = threadIdx.x;
    for (int j = 0; j < 7; ++j) {
        float s = 0.f;
        for (int q = t; q < 1024; q += 256) s += partial[(long)q * 7 + j];
        red[t] = s;
        __syncthreads();
        for (int off = 128; off > 0; off >>= 1) {
            if (t < off) red[t] += red[t + off];
            __syncthreads();
        }
        if (t == 0) tot[j] = red[0];
        __syncthreads();
    }
    if (t == 0) {
        float cnt = tot[0];
        float u1 = tot[1] / cnt, u2 = tot[3] / cnt, u3 = tot[5] / cnt;
        scal[0] = u1; scal[1] = sqrtf(tot[2] / cnt - u1 * u1);
        scal[2] = u2; scal[3] = sqrtf(tot[4] / cnt - u2 * u2);
        scal[4] = u3; scal[5] = sqrtf(tot[6] / cnt - u3 * u3);
    }
}

// ---------------------------------------------------------------------------
// Kernel 3: fused 8-channel separable conv (WMMA f32 16x16x4) + combine.
// One 32x32 output tile per 128-thread (4-wave) workgroup.
//
// WMMA fragment layouts (ISA 7.12.2, wave32):
//   A 16x4:  lane L, reg v -> (M = L&15, K = v + 2*(L>>4))
//   B 4x16:  lane L, reg v -> (K = v + 2*(L>>4), N = L&15)
//   C/D:     lane L, reg r -> (M = r + 8*(L>>4), N = L&15)
// ---------------------------------------------------------------------------
__global__ __launch_bounds__(128)
void ssim_blend(const float* __restrict__ img1, const float* __restrict__ img2,
                const float* __restrict__ img3, const int* __restrict__ mask,
                const float* __restrict__ scal, float* __restrict__ out) {
    extern __shared__ float lds[];
    float* INa = lds;
    float* INb = INa + IN_SZ;
    float* INc = INb + IN_SZ;
    float* HB  = INc + IN_SZ;     // 4 * HB_SZ (per channel-group)

    const int tid  = threadIdx.x;
    const int lane = tid & 31;
    const int wave = tid >> 5;
    const int m16   = lane & 15;
    const int half2 = (lane >> 4) << 1;   // K offset 0/2
    const int half8 = (lane >> 4) << 3;   // M offset 0/8 (C/D)
    const int tile_r0 = blockIdx.y * OT;
    const int tile_c0 = blockIdx.x * OT;

    // ---- async tile load: valid cells via ASYNC pipe, pad cells zeroed -----
    // (disjoint address sets -> no DS/ASYNC completion-order hazard)
    const unsigned ldsbase = __builtin_amdgcn_groupstaticsize();
    const unsigned ldsA = ldsbase;
    const unsigned ldsB = ldsbase + (unsigned)IN_SZ * 4u;
    const unsigned ldsC = ldsbase + (unsigned)IN_SZ * 8u;

#pragma unroll 1
    for (int it = 0; it < (IN_SZ + 127) / 128; ++it) {
        int i = tid + it * 128;
        if (i < IN_SZ) {
            int r = i / IN_STRIDE, c = i - r * IN_STRIDE;
            int gr = tile_r0 - PAD + r, gc = tile_c0 - PAD + c;
            bool ok = (r < IT) && (c < IT) && (gr >= 0) && (gr < H_IMG) &&
                      (gc >= 0) && (gc < W_IMG);
            if (ok) {
                long idx = (long)gr * W_IMG + gc;
                unsigned d = (unsigned)i * 4u;
                unsigned long long pa = (unsigned long long)(uintptr_t)(img1 + idx);
                unsigned long long pb = (unsigned long long)(uintptr_t)(img2 + idx);
                unsigned long long pc = (unsigned long long)(uintptr_t)(img3 + idx);
                unsigned dA = ldsA + d, dB = ldsB + d, dC = ldsC + d;
                asm volatile("global_load_async_to_lds_b32 %0, %1, off"
                             :: "v"(dA), "v"(pa) : "memory");
                asm volatile("global_load_async_to_lds_b32 %0, %1, off"
                             :: "v"(dB), "v"(pb) : "memory");
                asm volatile("global_load_async_to_lds_b32 %0, %1, off"
                             :: "v"(dC), "v"(pc) : "memory");
            } else {
                INa[i] = 0.f; INb[i] = 0.f; INc[i] = 0.f;
            }
        }
    }
    asm volatile("s_wait_asynccnt 0" ::: "memory");
    __syncthreads();

    // ---- per-lane Toeplitz weights: d = (4*kci + v + half2) - m16 ----------
    // Same values serve horizontal B-fragments and vertical A-fragments.
    float w0[7], w1[7];
#pragma unroll
    for (int kci = 0; kci < 7; ++kci) {
        int d0 = kci * 4 + half2 - m16;
        int d1 = d0 + 1;
        w0[kci] = (d0 >= 0 && d0 <= 10) ? gweight(d0) : 0.f;
        w1[kci] = (d1 >= 0 && d1 <= 10) ? gweight(d1) : 0.f;
    }

    v8f acc[8];
#pragma unroll
    for (int ch = 0; ch < 8; ++ch) acc[ch] = v8f_zero();

    const int vr0 = (wave >> 1) * 16;   // this wave's output 16x16 tile
    const int vc0 = (wave & 1) * 16;

#pragma unroll
    for (int grp = 0; grp < 2; ++grp) {
        // ---- horizontal: H[r][j] = sum_k X[r][k] * g[k-j]; 6 jobs/wave ----
#pragma unroll
        for (int q = 0; q < 6; ++q) {
            const int hj   = wave * 6 + q;
            const int chl  = hj & 3;
            const int tile = hj >> 2;
            const int hr0  = (tile >> 1) * 16;
            const int hc0  = (tile & 1) * 16;
            const int ch   = grp * 4 + chl;            // wave-uniform
            const int roff = (hr0 + m16) * IN_STRIDE;
            v8f hacc = v8f_zero();
#pragma unroll
            for (int kci = 0; kci < 7; ++kci) {
                const int col0 = hc0 + kci * 4 + half2;   // even -> 8B aligned
                v2f av = *(const v2f*)&INa[roff + col0];
                v2f bv = *(const v2f*)&INb[roff + col0];
                v2f cv = *(const v2f*)&INc[roff + col0];
                v2f afrag, bfrag;
#pragma unroll
                for (int v = 0; v < 2; ++v) {
                    // branchless channel value: pure v_cndmask selects
                    float f1 = (ch == 2 || ch == 5) ? cv[v]
                             : ((ch == 1 || ch == 4 || ch == 7) ? bv[v] : av[v]);
                    float f2 = (ch < 3) ? 1.f
                             : ((ch == 3) ? av[v] : ((ch == 4) ? bv[v] : cv[v]));
                    afrag[v] = f1 * f2;
                }
                bfrag[0] = w0[kci];
                bfrag[1] = w1[kci];
                hacc = __builtin_amdgcn_wmma_f32_16x16x4_f32(
                    false, afrag, false, bfrag, (short)0, hacc, false, false);
            }
#pragma unroll
            for (int r = 0; r < 8; ++r) {
                int row = hr0 + r + half8;
                HB[chl * HB_SZ + row * HB_STRIDE + hc0 + m16] = hacc[r];
            }
        }
        __syncthreads();

        // ---- vertical: Y[i][j] = sum_k g[k-i] * H[vr0+k][vc0+j] -----------
#pragma unroll
        for (int chl = 0; chl < 4; ++chl) {
            const float* Hc = &HB[chl * HB_SZ];
            v8f vacc = acc[grp * 4 + chl];
#pragma unroll
            for (int kci = 0; kci < 7; ++kci) {
                const int kg0 = kci * 4 + half2;
                v2f afrag, bfrag;
                afrag[0] = w0[kci];
                afrag[1] = w1[kci];
                bfrag[0] = Hc[(vr0 + kg0)     * HB_STRIDE + vc0 + m16];
                bfrag[1] = Hc[(vr0 + kg0 + 1) * HB_STRIDE + vc0 + m16];
                vacc = __builtin_amdgcn_wmma_f32_16x16x4_f32(
                    false, afrag, false, bfrag, (short)0, vacc, false, false);
            }
            acc[grp * 4 + chl] = vacc;
        }
        __syncthreads();   // HB reused by next group
    }

    // ---- fused combine (8 conv results live in C/D fragments) --------------
    const float u1 = scal[0], s1 = scal[1];
    const float u2 = scal[2], s2 = scal[3];
    const float u3 = scal[4], s3 = scal[5];
#pragma unroll
    for (int r = 0; r < 8; ++r) {
        int lr = vr0 + r + half8;
        int lc = vc0 + m16;
        float Ea  = acc[0][r], Eb  = acc[1][r], Ec  = acc[2][r];
        float Ea2 = acc[3][r], Eb2 = acc[4][r], Ec2 = acc[5][r];
        float Eac = acc[6][r], Ebc = acc[7][r];
        float va = Ea2 - Ea * Ea;
        float vb = Eb2 - Eb * Eb;
        float vc = Ec2 - Ec * Ec;
        float gamma2 = (Eac - Ea * Ec) / sqrtf(va) / sqrtf(vc);
        float gamma1 = (Ebc - Eb * Ec) / sqrtf(vb) / sqrtf(vc);
        long gidx = (long)(tile_r0 + lr) * W_IMG + (tile_c0 + lc);
        bool msk = mask[gidx] > 0;
        if (!msk) gamma1 = 0.f;
        float g1s   = gamma1 * gamma1;
        float denom = 1.f - g1s;
        float phi1  = g1s * (1.f - gamma2) / denom;
        float phi2  = (gamma2 - g1s) / denom;
        float a = INa[(lr + PAD) * IN_STRIDE + (lc + PAD)];
        float b = INb[(lr + PAD) * IN_STRIDE + (lc + PAD)];
        float n1 = (a - u1) / s1;
        float n2 = (b - u2) / s2;
        float val = u3 + (n1 * phi2 + n2 * phi1) * s3;
        out[gidx] = msk ? val : -15.f;
    }
}

// ---------------------------------------------------------------------------
extern "C" void kernel_launch(void* const* d_in, const int* in_sizes, int n_in,
                              void* d_out, int out_size, void* d_ws, size_t ws_size,
                              hipStream_t stream) {
    const float* img1 = (const float*)d_in[0];
    const float* img2 = (const float*)d_in[1];
    const float* img3 = (const float*)d_in[2];
    const int*   mask = (const int*)d_in[3];
    float* wsF     = (float*)d_ws;
    float* scal    = wsF;        // 8 floats reserved
    float* partial = wsF + 8;    // 1024 * 7 floats

    reduce_partial<<<1024, 256, 0, stream>>>(img1, img2, img3, mask, partial);
    reduce_final<<<1, 256, 0, stream>>>(partial, scal);

    dim3 grid(W_IMG / OT, H_IMG / OT);   // 64 x 64 tiles
    size_t shmem = (size_t)(3 * IN_SZ + 4 * HB_SZ) * sizeof(float);  // 60 KB
    ssim_blend<<<grid, 128, shmem, stream>>>(img1, img2, img3, mask, scal,
                                             (float*)d_out);
}